// mix_model2_5540507812308
// MI455X (gfx1250) — compile-verified
//
#include <hip/hip_runtime.h>
#include <hip/hip_bf16.h>

typedef __attribute__((ext_vector_type(2))) float v2f;
typedef __attribute__((ext_vector_type(8))) float v8f;

#define NODE_F 102   // 96 + 6
#define D_IN   96
#define FAN_IN 300   // 102 + 102 + 96
#define OUT_N  96
#define KSTEPS 75    // 300 / 4

// ---------------------------------------------------------------------------
// 1) zero node accumulator region of d_out and the denominator array
// ---------------------------------------------------------------------------
__global__ void k_zero(float* __restrict__ node, float* __restrict__ den, int N) {
    long t  = (long)blockIdx.x * blockDim.x + threadIdx.x;
    long nn = (long)N * NODE_F;
    if (t < nn)          node[t] = 0.0f;
    else if (t < nn + N) den[t - nn] = 0.0f;
}

// ---------------------------------------------------------------------------
// 2) masked scatter-sum: one 96-thread block per edge
// ---------------------------------------------------------------------------
__global__ void k_scatter(const float* __restrict__ eattr, const float* __restrict__ mask,
                          const int* __restrict__ eidx, float* __restrict__ node,
                          float* __restrict__ den, int E) {
    int e = blockIdx.x;
    int d = threadIdx.x;                  // 0..95
    float m = mask[e];
    int   c = eidx[E + e];                // col = edge_index[1][e]
    unsafeAtomicAdd(node + (size_t)c * NODE_F + d, eattr[(size_t)e * D_IN + d] * m);
    if (d == 0) unsafeAtomicAdd(den + c, m);
}

// ---------------------------------------------------------------------------
// 3) node_rep = num/(den+1) ; append x
// ---------------------------------------------------------------------------
__global__ void k_finalize(const float* __restrict__ x, const float* __restrict__ den,
                           float* __restrict__ node, int N) {
    long t = (long)blockIdx.x * blockDim.x + threadIdx.x;
    if (t >= (long)N * NODE_F) return;
    int n = (int)(t / NODE_F);
    int c = (int)(t % NODE_F);
    if (c < D_IN) node[t] = node[t] / (den[n] + 1.0f);
    else          node[t] = x[(size_t)n * 6 + (c - D_IN)];
}

// ---------------------------------------------------------------------------
// 4) tail mixing (both change paths), snapshot semantics for duplicate idxs
// ---------------------------------------------------------------------------
__global__ void k_tailmix(const int* __restrict__ chg, const int* __restrict__ sup,
                          const int* __restrict__ tails, int ntails,
                          const float* __restrict__ stail, float* __restrict__ node) {
    __shared__ float snap[128 * NODE_F];
    int c      = threadIdx.x;            // block of 128, columns 0..101 active
    int change = chg[0];
    int is_sup = sup[0];
    if (change && c < NODE_F) {
        for (int i = 0; i < ntails; ++i)
            snap[i * NODE_F + c] = node[(size_t)tails[i] * NODE_F + c];
    }
    __syncthreads();
    if (!change || c >= NODE_F) return;
    if (is_sup) {
        float s = 0.0f;
        for (int i = 0; i < ntails; ++i) s += snap[i * NODE_F + c];
        float mv = s / (float)ntails;
        for (int i = 0; i < ntails; ++i) node[(size_t)tails[i] * NODE_F + c] = mv;
    } else {
        float sv = stail[c];
        for (int i = 0; i < ntails; ++i)
            node[(size_t)tails[i] * NODE_F + c] = 0.1f * sv + 0.9f * snap[i * NODE_F + c];
    }
}

// ---------------------------------------------------------------------------
// 5) gather-GEMM with f32 WMMA: per wave, 32 edges x 96 outputs
// ---------------------------------------------------------------------------
__device__ __forceinline__ void wmma_step(const float* __restrict__ a0p,
                                          const float* __restrict__ a1p,
                                          const float* __restrict__ wlbase,
                                          int lm, v8f acc[2][6]) {
    v2f a0 = *(const v2f*)a0p;
    v2f a1 = *(const v2f*)a1p;
    const v2f* wb = (const v2f*)wlbase;
    #pragma unroll
    for (int j = 0; j < 6; ++j) {
        v2f b = wb[j * 16 + lm];
        acc[0][j] = __builtin_amdgcn_wmma_f32_16x16x4_f32(
            false, a0, false, b, (short)0, acc[0][j], false, false);
        acc[1][j] = __builtin_amdgcn_wmma_f32_16x16x4_f32(
            false, a1, false, b, (short)0, acc[1][j], false, false);
    }
}

__global__ __launch_bounds__(256)
void k_edge_gemm(const float* __restrict__ node,   // [N,102]
                 const int*   __restrict__ eidx,   // [2,E]
                 const float* __restrict__ eattr,  // [E,96]
                 const float* __restrict__ W,      // [96,300] row-major
                 const float* __restrict__ bias,   // [96]
                 float* __restrict__ out,          // [E,96]
                 int E) {
    extern __shared__ float wl[];                  // 300*96 floats, swizzled

    // Stage W into LDS, pre-swizzled: for k-step s, half h (K=4s+2h), column n,
    // store pair (W^T[k][n], W^T[k+1][n]) contiguously -> one ds_load_b64 per B frag.
    for (int idx = threadIdx.x; idx < FAN_IN * OUT_N; idx += 256) {
        int k = idx / OUT_N, n = idx % OUT_N;
        int s = k >> 2, h = (k >> 1) & 1, p = k & 1;
        wl[(((s * 2 + h) * OUT_N) + n) * 2 + p] = W[(size_t)n * FAN_IN + k];
    }
    __syncthreads();

    const int lane = threadIdx.x & 31;
    const int wv   = threadIdx.x >> 5;
    const int lm   = lane & 15;
    const int lh   = lane >> 4;
    const int tb   = (blockIdx.x * 8 + wv) * 32;   // 32 edges per wave

    int e0 = tb + lm, e1 = tb + 16 + lm;
    int e0c = e0 < E ? e0 : E - 1;
    int e1c = e1 < E ? e1 : E - 1;
    int r0 = eidx[e0c], c0 = eidx[E + e0c];
    int r1 = eidx[e1c], c1 = eidx[E + e1c];

    v8f acc[2][6];
    #pragma unroll
    for (int j = 0; j < 6; ++j) {
        float bv = bias[j * 16 + lm];              // bias depends only on N column
        #pragma unroll
        for (int r2 = 0; r2 < 8; ++r2) { acc[0][j][r2] = bv; acc[1][j][r2] = bv; }
    }

    // f = 4s + 2*lh walks the 300-wide concat [row(102) | col(102) | eattr(96)].
    // Segment crossings: row->col is lane-dependent only at s==25; col->eattr
    // is uniform at s==51.  Three branch-free strided ranges + 1 mixed step.
    const float* base_r0 = node  + (size_t)r0  * NODE_F + 2 * lh;
    const float* base_r1 = node  + (size_t)r1  * NODE_F + 2 * lh;
    const float* base_c0 = node  + (size_t)c0  * NODE_F + 2 * lh - NODE_F;
    const float* base_c1 = node  + (size_t)c1  * NODE_F + 2 * lh - NODE_F;
    const float* base_e0 = eattr + (size_t)e0c * D_IN   + 2 * lh - 2 * NODE_F;
    const float* base_e1 = eattr + (size_t)e1c * D_IN   + 2 * lh - 2 * NODE_F;
    const float* wl0     = wl + lh * (2 * OUT_N);  // + s*384 floats per step

    #pragma unroll 1
    for (int s = 0; s < 25; ++s)
        wmma_step(base_r0 + 4 * s, base_r1 + 4 * s, wl0 + 384 * s, lm, acc);

    {   // s == 25: lh=0 reads row[100:102], lh=1 reads col[0:2]
        const float* q0 = lh ? (node + (size_t)c0 * NODE_F)
                             : (node + (size_t)r0 * NODE_F + 100);
        const float* q1 = lh ? (node + (size_t)c1 * NODE_F)
                             : (node + (size_t)r1 * NODE_F + 100);
        wmma_step(q0, q1, wl0 + 384 * 25, lm, acc);
    }

    #pragma unroll 1
    for (int s = 26; s < 51; ++s)
        wmma_step(base_c0 + 4 * s, base_c1 + 4 * s, wl0 + 384 * s, lm, acc);

    #pragma unroll 1
    for (int s = 51; s < KSTEPS; ++s)
        wmma_step(base_e0 + 4 * s, base_e1 + 4 * s, wl0 + 384 * s, lm, acc);

    // C/D layout: VGPR r -> M = r + 8*(lane>>4), N = lane&15
    #pragma unroll
    for (int m = 0; m < 2; ++m) {
        #pragma unroll
        for (int j = 0; j < 6; ++j) {
            #pragma unroll
            for (int r2 = 0; r2 < 8; ++r2) {
                int edge = tb + m * 16 + r2 + 8 * lh;
                if (edge < E) out[(size_t)edge * OUT_N + j * 16 + lm] = acc[m][j][r2];
            }
        }
    }
}

// ---------------------------------------------------------------------------
extern "C" void kernel_launch(void* const* d_in, const int* in_sizes, int n_in,
                              void* d_out, int out_size, void* d_ws, size_t ws_size,
                              hipStream_t stream) {
    const int*   change     = (const int*)d_in[0];
    const int*   is_support = (const int*)d_in[1];
    const int*   tails      = (const int*)d_in[3];
    const float* x          = (const float*)d_in[4];
    const int*   eidx       = (const int*)d_in[5];
    const float* eattr      = (const float*)d_in[6];
    const float* mask       = (const float*)d_in[7];
    const float* stail      = (const float*)d_in[9];
    const float* W          = (const float*)d_in[10];
    const float* bias       = (const float*)d_in[11];

    int N      = in_sizes[4] / 6;
    int E      = in_sizes[7];
    int ntails = in_sizes[3];
    if (ntails > 128) ntails = 128;

    float* node     = (float*)d_out;                          // [N,102]
    float* edge_out = (float*)d_out + (size_t)N * NODE_F;     // [E,96]
    float* den      = (float*)d_ws;                           // [N]

    long ztot = (long)N * (NODE_F + 1);
    k_zero<<<(unsigned)((ztot + 255) / 256), 256, 0, stream>>>(node, den, N);
    k_scatter<<<(unsigned)E, D_IN, 0, stream>>>(eattr, mask, eidx, node, den, E);
    k_finalize<<<(unsigned)(((long)N * NODE_F + 255) / 256), 256, 0, stream>>>(x, den, node, N);
    k_tailmix<<<1, 128, 0, stream>>>(change, is_support, tails, ntails, stail, node);

    const int ldsW = FAN_IN * OUT_N * (int)sizeof(float);     // 115200 B
    (void)hipFuncSetAttribute((const void*)k_edge_gemm,
                              hipFuncAttributeMaxDynamicSharedMemorySize, ldsW);
    k_edge_gemm<<<(unsigned)((E + 255) / 256), 256, ldsW, stream>>>(
        node, eidx, eattr, W, bias, edge_out, E);
}